// PatchEmbedding2_34308198760639
// MI455X (gfx1250) — compile-verified
//
#include <hip/hip_runtime.h>
#include <hip/hip_bf16.h>
#include <math.h>

// ---------------------------------------------------------------------------
// PatchEmbedding on MI455X (gfx1250): fp32 WMMA 16x16x4 GEMM over gathered
// variable-size patches.  B=8, C=3, IMG=1024, PS=32, E=768, NP=992.
// ---------------------------------------------------------------------------

typedef float v2f __attribute__((ext_vector_type(2)));
typedef float v8f __attribute__((ext_vector_type(8)));

#define EMB      768
#define IMGSZ    1024
#define PS       32
#define K32      3072          // 3 * 32 * 32
#define LDSTRIDE 68            // 16-row A tile stride (floats): 4*r+k banks, 16B aligned
#define USE_ASYNC_A 1          // gfx1250 global_load_async_to_lds_b128 for A gather

// ---------------------------------------------------------------------------
// Kernel 1: token layout (single block of 1024 threads).
// ---------------------------------------------------------------------------
__global__ __launch_bounds__(1024) void tok_layout_kernel(
    const float* __restrict__ iw, int NP,
    int* __restrict__ tokRow, int* __restrict__ tokCol,
    int* __restrict__ classList, int* __restrict__ classCount, int Tmax)
{
  const int tid = threadIdx.x;
  __shared__ double sd[1024];
  __shared__ int    si[1024];

  sd[tid] = (tid < NP) ? (double)iw[tid] : 0.0;
  __syncthreads();
  for (int off = 512; off > 0; off >>= 1) {
    if (tid < off) sd[tid] += sd[tid + off];
    __syncthreads();
  }
  const double avg = sd[0] / (double)NP;

  int sz = PS, tp = 0, cls = -1;
  if (tid < NP) {
    double val = (double)PS * pow(avg, 0.05) / pow((double)iw[tid], 0.05);
    if (val > (double)PS) val = (double)PS;
    const double ch[4] = {32.0, 16.0, 8.0, 4.0};
    int best = 0; double bd = fabs(val - ch[0]);
    for (int c = 1; c < 4; ++c) {
      double d = fabs(val - ch[c]);
      if (d < bd) { bd = d; best = c; }   // ties keep first (larger size)
    }
    cls = best;
    sz  = PS >> best;                      // 32,16,8,4
    tp  = (sz == PS) ? 1 : (PS / sz);
  }

  // emission-order token start
  si[tid] = tp;
  __syncthreads();
  int incl = tp;
  for (int off = 1; off < 1024; off <<= 1) {
    int a = si[tid];
    int b = (tid >= off) ? si[tid - off] : 0;
    __syncthreads();
    incl = a + b;
    si[tid] = incl;
    __syncthreads();
  }
  const int tstart = incl - tp;

  // deterministic per-class compaction
  for (int c = 0; c < 4; ++c) {
    int contrib = (cls == c) ? tp : 0;
    si[tid] = contrib;
    __syncthreads();
    int cincl = contrib;
    for (int off = 1; off < 1024; off <<= 1) {
      int a = si[tid];
      int b = (tid >= off) ? si[tid - off] : 0;
      __syncthreads();
      cincl = a + b;
      si[tid] = cincl;
      __syncthreads();
    }
    if (cls == c && tid < NP) {
      const int cstart  = cincl - contrib;
      const int i       = tid;
      const int ini_col = (i * PS) % IMGSZ;
      const int ini_row = (i * PS * PS) / IMGSZ;
      for (int j = 0; j < tp; ++j) {
        int t = tstart + j, r, cc;
        if (sz == PS) { r = ini_row; cc = ini_col; }
        else {
          int ns = PS / sz;
          r  = ini_row + (j * sz) / ns;
          cc = ini_col + (j * sz) % (ns * sz);
        }
        tokRow[t] = r;
        tokCol[t] = cc;
        classList[c * Tmax + cstart + j] = t;
      }
    }
    if (tid == 1023) classCount[c] = si[1023];
    __syncthreads();
  }
}

// ---------------------------------------------------------------------------
// Kernel 2: folded weights in quad-K interleaved layout:
//   Wq[((k>>2)*EMB + e)*4 + (k&3)]
// so a WMMA B-fragment (two adjacent K per lane-half) is one 8B load, and the
// 32 lanes of a wave cover one contiguous 256B region.
// Folding matches reference reshape (E,C,kd,kd,s,s).sum(axis=(2,3)).
// Rows k in [Ks,Kpad) zero-filled.
// ---------------------------------------------------------------------------
__global__ __launch_bounds__(256) void fold_weights_kernel(
    const float* __restrict__ W, float* __restrict__ Wq,
    int s, int Ks, int Kpad)
{
  int idx = blockIdx.x * 256 + threadIdx.x;
  if (idx >= Kpad * EMB) return;
  int k = idx / EMB;
  int e = idx - k * EMB;
  float sum = 0.f;
  if (k < Ks) {
    int ss  = s * s;
    int chn = k / ss;
    int rem = k - chn * ss;
    int p   = rem / s;
    int q   = rem - p * s;
    int kd  = PS / s;
    int strideA = (PS * PS) / kd;   // 1024/kd
    const float* base = W + (size_t)e * K32 + chn * (PS * PS) + p * s + q;
    for (int a = 0; a < kd; ++a)
      for (int b = 0; b < kd; ++b)
        sum += base[a * strideA + b * ss];
  }
  Wq[((size_t)(k >> 2) * EMB + e) * 4 + (k & 3)] = sum;
}

// ---------------------------------------------------------------------------
// Kernel 3: per-size-class GEMM.  Block = 256 threads (8 wave32).
// Block computes a 16-row M-tile (rows = classPos*B + batch) x all 768 embeds.
// K streamed in 64-wide chunks: A async-gathered to LDS, B from quad-K Wq,
// accumulate with V_WMMA_F32_16X16X4_F32 (6 independent chains per wave).
// ---------------------------------------------------------------------------
__global__ __launch_bounds__(256) void patch_gemm_kernel(
    const float* __restrict__ x,          // (B,3,1024,1024)
    const float* __restrict__ Wq,         // (Kpad/4, 768, 4), zero-padded
    const float* __restrict__ bias,       // (768)
    const int*   __restrict__ classList,  // token ids for this class
    const int*   __restrict__ classCountPtr,
    const int*   __restrict__ tokRow,
    const int*   __restrict__ tokCol,
    float* __restrict__ out,              // (B, T, 768)
    int T, int Bn, int s, int Ks, int Kpad)
{
  const int count = *classCountPtr;
  const int tile  = blockIdx.x;
  if (tile * 16 >= count * Bn) return;    // uniform exit, before any barrier

  const int tid  = threadIdx.x;
  const int lane = tid & 31;
  const int wave = tid >> 5;

  __shared__ float As[16 * LDSTRIDE];     // ONLY LDS object -> LDS offset 0

  // ---- A-gather setup: thread owns (row ar, 4 k's starting at akb) --------
  const int ar  = tid & 15;
  const int akb = (tid >> 4) * 4;
  const int m0  = tile * 16 + ar;
  const int posA = m0 / Bn;
  const int bA   = m0 - posA * Bn;
  const bool rowValid = (posA < count);
  const float* xbase = x;
  int prow = 0, pcol = 0;
  if (rowValid) {
    int t = classList[posA];
    prow  = tokRow[t];
    pcol  = tokCol[t];
    xbase = x + (size_t)bA * 3 * IMGSZ * IMGSZ;
  }
  const int ss = s * s;
  const unsigned ldsByteOff = (unsigned)((ar * LDSTRIDE + akb) * 4);

  v8f acc[6];
  {
    v8f z = {};
    #pragma unroll
    for (int j = 0; j < 6; ++j) acc[j] = z;
  }

  const int arow  = lane & 15;          // A fragment: M index
  const int akoff = (lane >> 4) * 2;    // A/B fragment: K sub-offset per lane half
  const int eLane = (lane & 15);

  for (int kc = 0; kc < Kpad; kc += 64) {
    __syncthreads();                    // previous chunk fully consumed
    // ---- gather 16B of patch pixels into LDS (dc%4==0 -> 16B aligned) ----
    const int k = kc + akb;
    if (rowValid && k < Ks) {
      int chn = k / ss;
      int rem = k - chn * ss;
      int dr  = rem / s;
      int dc  = rem - dr * s;
      const float* p = xbase + ((size_t)chn * IMGSZ + (prow + dr)) * IMGSZ + (pcol + dc);
#if USE_ASYNC_A
      // gfx1250 async global->LDS DMA, tracked by ASYNCcnt
      asm volatile("global_load_async_to_lds_b128 %0, %1, off"
                   :: "v"(ldsByteOff), "v"(p)
                   : "memory");
#else
      *reinterpret_cast<float4*>(&As[ar * LDSTRIDE + akb]) =
          *reinterpret_cast<const float4*>(p);
#endif
    } else {
      float4 z = make_float4(0.f, 0.f, 0.f, 0.f);
      *reinterpret_cast<float4*>(&As[ar * LDSTRIDE + akb]) = z;
    }
#if USE_ASYNC_A
    asm volatile("s_wait_asynccnt 0" ::: "memory");
#endif
    __syncthreads();

    // preload 16 A fragments (pairs merge into ds_load_2addr_b64)
    v2f afrag[16];
    #pragma unroll
    for (int ks = 0; ks < 16; ++ks) {
      const float* ap = &As[arow * LDSTRIDE + ks * 4 + akoff];
      afrag[ks].x = ap[0];
      afrag[ks].y = ap[1];
    }

    // ks-outer / j-inner: 6 independent accumulation chains, A reused 6x
    const float* wchunk = Wq + ((size_t)(kc >> 2) * EMB) * 4 + akoff;
    #pragma unroll
    for (int ks = 0; ks < 16; ++ks) {
      const float* wk = wchunk + (size_t)ks * EMB * 4;
      #pragma unroll
      for (int j = 0; j < 6; ++j) {
        const int e = (wave * 6 + j) * 16 + eLane;
        v2f bfrag = *reinterpret_cast<const v2f*>(wk + (size_t)e * 4);
        acc[j] = __builtin_amdgcn_wmma_f32_16x16x4_f32(
            false, afrag[ks], false, bfrag, (short)0, acc[j], false, false);
      }
    }
  }

  // ---- bias + store (C layout: VGPR v -> M = v + 8*(lane>=16), N = lane&15)
  #pragma unroll
  for (int j = 0; j < 6; ++j) {
    const int e = (wave * 6 + j) * 16 + eLane;
    const float bv = bias[e];
    #pragma unroll
    for (int v = 0; v < 8; ++v) {
      const int mrow = v + ((lane >> 4) << 3);
      const int mm   = tile * 16 + mrow;
      const int pos  = mm / Bn;
      const int bb   = mm - pos * Bn;
      if (pos < count) {
        const int t = classList[pos];
        out[((size_t)bb * T + t) * EMB + e] = acc[j][v] + bv;
      }
    }
  }
}

// ---------------------------------------------------------------------------
// Host launcher
// ---------------------------------------------------------------------------
extern "C" void kernel_launch(void* const* d_in, const int* in_sizes, int n_in,
                              void* d_out, int out_size, void* d_ws, size_t ws_size,
                              hipStream_t stream) {
  (void)n_in; (void)ws_size;
  const float* x    = (const float*)d_in[0];
  const float* iw   = (const float*)d_in[1];
  const float* W    = (const float*)d_in[2];
  const float* bias = (const float*)d_in[3];
  float* out = (float*)d_out;

  const int NP   = in_sizes[1];                          // 992
  const int Bn   = in_sizes[0] / (3 * IMGSZ * IMGSZ);    // 8
  const int T    = out_size / (Bn * EMB);                // 995
  const int Tmax = NP * 8;                               // worst case (all s=4)

  // workspace carve-up (~12.8 MB total)
  char* ws = (char*)d_ws;
  int* classCount = (int*)ws;  ws += 64;
  int* tokRow     = (int*)ws;  ws += (size_t)Tmax * 4;
  int* tokCol     = (int*)ws;  ws += (size_t)Tmax * 4;
  int* classList  = (int*)ws;  ws += (size_t)4 * Tmax * 4;
  float* Wq32 = (float*)ws;    ws += (size_t)3072 * EMB * 4;
  float* Wq16 = (float*)ws;    ws += (size_t)768  * EMB * 4;
  float* Wq8  = (float*)ws;    ws += (size_t)192  * EMB * 4;
  float* Wq4  = (float*)ws;    ws += (size_t)64   * EMB * 4;

  tok_layout_kernel<<<1, 1024, 0, stream>>>(iw, NP, tokRow, tokCol,
                                            classList, classCount, Tmax);

  struct Cls { float* wq; int s, Ks, Kpad; };
  const Cls cl[4] = { {Wq32, 32, 3072, 3072},
                      {Wq16, 16,  768,  768},
                      {Wq8,   8,  192,  192},
                      {Wq4,   4,   48,   64} };

  for (int c = 0; c < 4; ++c) {
    int n = cl[c].Kpad * EMB;
    fold_weights_kernel<<<(n + 255) / 256, 256, 0, stream>>>(
        W, cl[c].wq, cl[c].s, cl[c].Ks, cl[c].Kpad);
  }

  const int maxTiles = (T * Bn + 15) / 16;   // over-launch; blocks exit on count
  for (int c = 0; c < 4; ++c) {
    patch_gemm_kernel<<<maxTiles, 256, 0, stream>>>(
        x, cl[c].wq, bias, classList + (size_t)c * Tmax, classCount + c,
        tokRow, tokCol, out, T, Bn, cl[c].s, cl[c].Ks, cl[c].Kpad);
  }
}